// GATEncoder_67173288509658
// MI455X (gfx1250) — compile-verified
//
#include <hip/hip_runtime.h>
#include <hip/hip_bf16.h>
#include <math.h>
#include <stdint.h>

typedef __attribute__((ext_vector_type(2))) float v2f;
typedef __attribute__((ext_vector_type(8))) float v8f;

#define NNODES 50000
#define NEDGES 800000
#define ETOT   (NEDGES + NNODES)
#define IN1    22
#define IN1P   24              // padded to multiple of 4
#define HID    64
#define HEADS  4
#define F1     (HEADS * HID)   // 256
#define OUT2   32
#define NEG_SLOPE 0.2f
#define BN_EPS 1e-5f

// ---------------------------------------------------------------------------
// Wave-level fp32 GEMM: each wave computes a 16 x (16*NT) slab of
// C[M][N] = A[M][K] * B, with B supplied TRANSPOSED as BT[N][K].
// K compile-time, multiple of 4 (operands pre-padded) -> no guards.
// One-step software pipeline: loads for K-step k+1 are issued before the
// WMMAs of step k, so waits amortize over a whole WMMA group.
// A fragment: M = lane%16, K = 2*(lane>=16)+v  -> one b64 load.
// B fragment: N = lane%16, K = 2*(lane>=16)+v  -> one b64 load (BT row).
// ---------------------------------------------------------------------------
template<int K, int NT>
__global__ __launch_bounds__(256) void gemm_wmma_f32(
    const float* __restrict__ A, const float* __restrict__ BT,
    float* __restrict__ C, int M, int N)
{
    const int lane = threadIdx.x & 31;
    const int wave = blockIdx.x * (blockDim.x >> 5) + (threadIdx.x >> 5);
    const int ngrp = N / (16 * NT);
    const int mtile = M >> 4;
    if (wave >= mtile * ngrp) return;            // wave-uniform: EXEC all-1 at WMMA
    const int tm = wave / ngrp;
    const int tg = wave - tm * ngrp;

    const int khalf = (lane >> 4) * 2;           // 0 or 2
    const float* arow = A + (size_t)(tm * 16 + (lane & 15)) * K + khalf;
    const float* brow[NT];
    v8f acc[NT];
#pragma unroll
    for (int j = 0; j < NT; ++j) {
        acc[j] = (v8f){};
        brow[j] = BT + (size_t)(tg * 16 * NT + j * 16 + (lane & 15)) * K + khalf;
    }

    // pipeline prologue: stage K-step 0
    v2f a_cur = *(const v2f*)(arow);
    v2f b_cur[NT];
#pragma unroll
    for (int j = 0; j < NT; ++j) b_cur[j] = *(const v2f*)(brow[j]);

#pragma unroll
    for (int k0 = 0; k0 < K; k0 += 4) {
        v2f a_nxt{};
        v2f b_nxt[NT];
        if (k0 + 4 < K) {                         // static (K compile-time)
            a_nxt = *(const v2f*)(arow + k0 + 4);
#pragma unroll
            for (int j = 0; j < NT; ++j)
                b_nxt[j] = *(const v2f*)(brow[j] + k0 + 4);
        }
#pragma unroll
        for (int j = 0; j < NT; ++j)
            acc[j] = __builtin_amdgcn_wmma_f32_16x16x4_f32(
                false, a_cur, false, b_cur[j], (short)0, acc[j], false, false);
        a_cur = a_nxt;
#pragma unroll
        for (int j = 0; j < NT; ++j) b_cur[j] = b_nxt[j];
    }

    const int rbase = tm * 16 + 8 * (lane >> 4); // D: M = r + 8*(lane>=16)
#pragma unroll
    for (int j = 0; j < NT; ++j) {
        const int col = tg * 16 * NT + j * 16 + (lane & 15);
#pragma unroll
        for (int r = 0; r < 8; ++r)
            C[(size_t)(rbase + r) * N + col] = acc[j][r];
    }
}

// ---------------------------------------------------------------------------
// Operand packing: xp[N][24] (zero-padded), w1t[256][24] (= W1^T padded),
// w2t[32][256] (= W2^T).
// ---------------------------------------------------------------------------
__global__ void pack_x_kernel(const float* __restrict__ x, float* __restrict__ xp)
{
    int t = blockIdx.x * blockDim.x + threadIdx.x;
    if (t >= NNODES * IN1P) return;
    int n = t / IN1P, c = t - n * IN1P;
    xp[t] = (c < IN1) ? x[(size_t)n * IN1 + c] : 0.0f;
}
__global__ void pack_w1t_kernel(const float* __restrict__ W1, float* __restrict__ w1t)
{
    int t = blockIdx.x * blockDim.x + threadIdx.x;
    if (t >= F1 * IN1P) return;
    int col = t / IN1P, k = t - col * IN1P;
    w1t[t] = (k < IN1) ? W1[(size_t)k * F1 + col] : 0.0f;
}
__global__ void pack_w2t_kernel(const float* __restrict__ W2, float* __restrict__ w2t)
{
    int t = blockIdx.x * blockDim.x + threadIdx.x;
    if (t >= OUT2 * F1) return;
    int col = t / F1, k = t - col * F1;
    w2t[t] = W2[(size_t)k * OUT2 + col];
}

// ---------------------------------------------------------------------------
// Attention dot products: as[n,h] = sum_c h[n,h,c]*att_src[h,c] (same for dst)
// ---------------------------------------------------------------------------
template<int HEADS_T, int CHEAD>
__global__ void att_dots_kernel(const float* __restrict__ h,
                                const float* __restrict__ att_src,
                                const float* __restrict__ att_dst,
                                float* __restrict__ a_src, float* __restrict__ a_dst)
{
    int t = blockIdx.x * blockDim.x + threadIdx.x;
    if (t >= NNODES * HEADS_T) return;
    int hd = t & (HEADS_T - 1);
    const float* hp = h + (size_t)t * CHEAD;            // [n][hd] contiguous
    const float* ws = att_src + hd * CHEAD;
    const float* wd = att_dst + hd * CHEAD;
    float ss = 0.f, sd = 0.f;
#pragma unroll 4
    for (int c = 0; c < CHEAD; ++c) { float v = hp[c]; ss += v * ws[c]; sd += v * wd[c]; }
    a_src[t] = ss; a_dst[t] = sd;
}

__global__ void fill_kernel(float* __restrict__ p, float v, int n)
{
    int t = blockIdx.x * blockDim.x + threadIdx.x;
    if (t < n) p[t] = v;
}

__device__ __forceinline__ void atomicMaxF(float* addr, float val)
{
    if (val >= 0.0f) atomicMax((int*)addr, __float_as_int(val));
    else             atomicMin((unsigned int*)addr, __float_as_uint(val));
}

__device__ __forceinline__ void edge_ends(const long long* __restrict__ ei, int e,
                                          int& s, int& d)
{
    if (e < NEDGES) { s = (int)ei[e]; d = (int)ei[NEDGES + e]; }
    else            { s = d = e - NEDGES; }
}

// ---------------------------------------------------------------------------
// Edge phase 1: logits + segment max.  One thread per (edge, head).
// ---------------------------------------------------------------------------
template<int HEADS_T>
__global__ void edge_max_kernel(const long long* __restrict__ ei,
                                const float* __restrict__ a_src,
                                const float* __restrict__ a_dst,
                                float* __restrict__ ebuf, float* __restrict__ segmax)
{
    int t = blockIdx.x * blockDim.x + threadIdx.x;
    if (t >= ETOT * HEADS_T) return;
    int e = t / HEADS_T, hd = t & (HEADS_T - 1);
    int s, d; edge_ends(ei, e, s, d);
    float v = a_src[s * HEADS_T + hd] + a_dst[d * HEADS_T + hd];
    v = (v > 0.0f) ? v : NEG_SLOPE * v;        // leaky_relu
    ebuf[t] = v;
    atomicMaxF(&segmax[d * HEADS_T + hd], v);
}

// ---------------------------------------------------------------------------
// Edge phase 2: segment sum of exp(e - m[dst]).
// ---------------------------------------------------------------------------
template<int HEADS_T>
__global__ void edge_sum_kernel(const long long* __restrict__ ei,
                                const float* __restrict__ ebuf,
                                const float* __restrict__ segmax,
                                float* __restrict__ segsum)
{
    int t = blockIdx.x * blockDim.x + threadIdx.x;
    if (t >= ETOT * HEADS_T) return;
    int e = t / HEADS_T, hd = t & (HEADS_T - 1);
    int s, d; edge_ends(ei, e, s, d); (void)s;
    float p = __expf(ebuf[t] - segmax[d * HEADS_T + hd]);
    atomicAdd(&segsum[d * HEADS_T + hd], p);
}

// ---------------------------------------------------------------------------
// Edge phase 3: weighted scatter-add of messages.  One wave per edge; lanes
// cover channels (coalesced), heads in outer loop.
// ---------------------------------------------------------------------------
template<int HEADS_T, int CHEAD>
__global__ __launch_bounds__(256) void edge_aggr_kernel(
    const long long* __restrict__ ei,
    const float* __restrict__ ebuf, const float* __restrict__ segmax,
    const float* __restrict__ segsum,
    const float* __restrict__ h, float* __restrict__ acc)
{
    int lane = threadIdx.x & 31;
    int e = blockIdx.x * (blockDim.x >> 5) + (threadIdx.x >> 5);
    if (e >= ETOT) return;
    int s, d; edge_ends(ei, e, s, d);
    const int F = HEADS_T * CHEAD;
    const float* hs = h + (size_t)s * F;
    float* ac = acc + (size_t)d * F;
#pragma unroll
    for (int hd = 0; hd < HEADS_T; ++hd) {
        float al = __expf(ebuf[(size_t)e * HEADS_T + hd] - segmax[d * HEADS_T + hd])
                 / (segsum[d * HEADS_T + hd] + 1e-16f);
#pragma unroll
        for (int c = lane; c < CHEAD; c += 32)
            atomicAdd(&ac[hd * CHEAD + c], al * hs[hd * CHEAD + c]);
    }
}

// ---------------------------------------------------------------------------
// out = ELU( BN( acc + bias ) )   (eval-mode BN)
// ---------------------------------------------------------------------------
__global__ void bias_bn_elu_kernel(const float* __restrict__ acc,
                                   const float* __restrict__ bias,
                                   const float* __restrict__ gamma,
                                   const float* __restrict__ beta,
                                   const float* __restrict__ mean,
                                   const float* __restrict__ var,
                                   float* __restrict__ out, int F)
{
    int t = blockIdx.x * blockDim.x + threadIdx.x;
    if (t >= NNODES * F) return;
    int c = t % F;
    float v = acc[t] + bias[c];
    v = (v - mean[c]) * rsqrtf(var[c] + BN_EPS) * gamma[c] + beta[c];
    out[t] = (v > 0.0f) ? v : (__expf(v) - 1.0f);
}

// ---------------------------------------------------------------------------
extern "C" void kernel_launch(void* const* d_in, const int* in_sizes, int n_in,
                              void* d_out, int out_size, void* d_ws, size_t ws_size,
                              hipStream_t stream)
{
    const float*     x        = (const float*)d_in[0];
    const long long* ei       = (const long long*)d_in[1];
    const float*     W1       = (const float*)d_in[2];
    const float*     att_src1 = (const float*)d_in[3];
    const float*     att_dst1 = (const float*)d_in[4];
    const float*     b1       = (const float*)d_in[5];
    const float*     W2       = (const float*)d_in[6];
    const float*     att_src2 = (const float*)d_in[7];
    const float*     att_dst2 = (const float*)d_in[8];
    const float*     b2       = (const float*)d_in[9];
    const float*     gamma1   = (const float*)d_in[10];
    const float*     beta1    = (const float*)d_in[11];
    const float*     mean1    = (const float*)d_in[12];
    const float*     var1     = (const float*)d_in[13];
    const float*     gamma2   = (const float*)d_in[14];
    const float*     beta2    = (const float*)d_in[15];
    const float*     mean2    = (const float*)d_in[16];
    const float*     var2     = (const float*)d_in[17];

    // workspace layout (floats; all offsets even -> 8B alignment for b64 loads)
    float* ws = (float*)d_ws;
    size_t o = 0;
    float* h1   = ws + o; o += (size_t)NNODES * F1;     // 12.8M
    float* acc1 = ws + o; o += (size_t)NNODES * F1;     // 12.8M (becomes hpost)
    float* xp   = ws + o; o += (size_t)NNODES * IN1P;   // 1.2M
    float* w1t  = ws + o; o += (size_t)F1 * IN1P;       // 6144
    float* w2t  = ws + o; o += (size_t)OUT2 * F1;       // 8192
    float* as1  = ws + o; o += (size_t)NNODES * HEADS;
    float* ad1  = ws + o; o += (size_t)NNODES * HEADS;
    float* m1   = ws + o; o += (size_t)NNODES * HEADS;
    float* s1   = ws + o; o += (size_t)NNODES * HEADS;
    float* eb   = ws + o; o += (size_t)ETOT * HEADS;    // 3.4M (reused for layer 2)
    float* h2   = ws + o; o += (size_t)NNODES * OUT2;
    float* acc2 = ws + o; o += (size_t)NNODES * OUT2;
    float* as2  = ws + o; o += (size_t)NNODES;
    float* ad2  = ws + o; o += (size_t)NNODES;
    float* m2   = ws + o; o += (size_t)NNODES;
    float* s2   = ws + o; o += (size_t)NNODES;
    float* eb2  = eb;

    const int TPB = 256;
    auto blks = [](long long n, int tpb) { return (int)((n + tpb - 1) / tpb); };

    // ===== operand packing =====
    pack_x_kernel  <<<blks((long long)NNODES * IN1P, TPB), TPB, 0, stream>>>(x, xp);
    pack_w1t_kernel<<<blks((long long)F1 * IN1P, TPB), TPB, 0, stream>>>(W1, w1t);
    pack_w2t_kernel<<<blks((long long)OUT2 * F1, TPB), TPB, 0, stream>>>(W2, w2t);

    // ===== Layer 1: 22 -> 4 x 64, concat =====
    {   // C[N][256] = xp[N][24] * W1 ; NT=4 -> 12500 waves
        int waves = (NNODES / 16) * (F1 / 64);
        gemm_wmma_f32<IN1P, 4><<<blks((long long)waves * 32, TPB), TPB, 0, stream>>>(
            xp, w1t, h1, NNODES, F1);
    }
    att_dots_kernel<HEADS, HID><<<blks((long long)NNODES * HEADS, TPB), TPB, 0, stream>>>(
        h1, att_src1, att_dst1, as1, ad1);
    fill_kernel<<<blks((long long)NNODES * HEADS, TPB), TPB, 0, stream>>>(m1, -INFINITY, NNODES * HEADS);
    fill_kernel<<<blks((long long)NNODES * HEADS, TPB), TPB, 0, stream>>>(s1, 0.0f, NNODES * HEADS);
    fill_kernel<<<blks((long long)NNODES * F1, TPB), TPB, 0, stream>>>(acc1, 0.0f, NNODES * F1);
    edge_max_kernel<HEADS><<<blks((long long)ETOT * HEADS, TPB), TPB, 0, stream>>>(ei, as1, ad1, eb, m1);
    edge_sum_kernel<HEADS><<<blks((long long)ETOT * HEADS, TPB), TPB, 0, stream>>>(ei, eb, m1, s1);
    edge_aggr_kernel<HEADS, HID><<<blks((long long)ETOT * 32, TPB), TPB, 0, stream>>>(
        ei, eb, m1, s1, h1, acc1);
    bias_bn_elu_kernel<<<blks((long long)NNODES * F1, TPB), TPB, 0, stream>>>(
        acc1, b1, gamma1, beta1, mean1, var1, acc1, F1);       // in-place -> hpost

    // ===== Layer 2: 256 -> 32, 1 head, mean (=identity) =====
    {   // C[N][32] = hpost[N][256] * W2 ; NT=2 -> 3125 waves
        int waves = (NNODES / 16) * (OUT2 / 32);
        gemm_wmma_f32<F1, 2><<<blks((long long)waves * 32, TPB), TPB, 0, stream>>>(
            acc1, w2t, h2, NNODES, OUT2);
    }
    att_dots_kernel<1, OUT2><<<blks((long long)NNODES, TPB), TPB, 0, stream>>>(
        h2, att_src2, att_dst2, as2, ad2);
    fill_kernel<<<blks((long long)NNODES, TPB), TPB, 0, stream>>>(m2, -INFINITY, NNODES);
    fill_kernel<<<blks((long long)NNODES, TPB), TPB, 0, stream>>>(s2, 0.0f, NNODES);
    fill_kernel<<<blks((long long)NNODES * OUT2, TPB), TPB, 0, stream>>>(acc2, 0.0f, NNODES * OUT2);
    edge_max_kernel<1><<<blks((long long)ETOT, TPB), TPB, 0, stream>>>(ei, as2, ad2, eb2, m2);
    edge_sum_kernel<1><<<blks((long long)ETOT, TPB), TPB, 0, stream>>>(ei, eb2, m2, s2);
    edge_aggr_kernel<1, OUT2><<<blks((long long)ETOT * 32, TPB), TPB, 0, stream>>>(
        ei, eb2, m2, s2, h2, acc2);
    bias_bn_elu_kernel<<<blks((long long)NNODES * OUT2, TPB), TPB, 0, stream>>>(
        acc2, b2, gamma2, beta2, mean2, var2, (float*)d_out, OUT2);
}